// DGCNN_GraphLayer_75290776699036
// MI455X (gfx1250) — compile-verified
//
#include <hip/hip_runtime.h>
#include <cstdint>

// ---------------------------------------------------------------------------
// DGCNN forward for MI455X (gfx1250, wave32).
// All GEMM-ish work uses V_WMMA_F32_16X16X4_F32 (full f32 precision; the
// network is memory/selection bound, ~100 GFLOP total, so low-precision WMMA
// would only add numeric risk, not speed). Inner loops are branch-free for
// the channel-aligned layers (2, 3, final); only layer 1 (Cin=3) uses the
// guarded generic path.
// ---------------------------------------------------------------------------

#define B_    8
#define N_    2048
#define KNN   20
#define NPW   8            // points per wave in edge-conv kernel
#define EPSV  1e-5f
#define INF_  3.402823466e+38f

typedef __attribute__((ext_vector_type(2))) float v2f;
typedef __attribute__((ext_vector_type(8))) float v8f;

static __device__ inline v8f wmma_f32_4(v2f a, v2f b, v8f c) {
  // D(16x16,f32) = A(16x4,f32) * B(4x16,f32) + C
  return __builtin_amdgcn_wmma_f32_16x16x4_f32(
      /*neg_a=*/false, a, /*neg_b=*/false, b,
      /*c_mod=*/(short)0, c, /*reuse_a=*/false, /*reuse_b=*/false);
}

// ---------------------------------------------------------------------------
// sq[b][n] = sum_c x[b][c][n]^2
// ---------------------------------------------------------------------------
__global__ void sq_kernel(const float* __restrict__ x, float* __restrict__ sq, int Cin) {
  int i = blockIdx.x * blockDim.x + threadIdx.x;
  if (i >= B_ * N_) return;
  int b = i / N_, n = i % N_;
  const float* xb = x + (size_t)b * Cin * N_;
  float s = 0.f;
  for (int c = 0; c < Cin; ++c) { float v = xb[(size_t)c * N_ + n]; s += v * v; }
  sq[i] = s;
}

// ---------------------------------------------------------------------------
// sdist[row][col] = sq[col] - 2*<x_row, x_col>   (same ranking as distance)
// One wave per 16x16 tile, WMMA f32 16x16x4, K loop over channels.
// ALIGNED: Cin % 4 == 0 -> branch-free loads.
// ---------------------------------------------------------------------------
template <bool ALIGNED>
__global__ void dist_kernel(const float* __restrict__ x, const float* __restrict__ sq,
                            float* __restrict__ sdist, int b, int Cin) {
  int lane = threadIdx.x;
  int l16  = lane & 15;
  bool hi  = lane >= 16;
  int row0 = blockIdx.y * 16, col0 = blockIdx.x * 16;
  const float* xb = x + (size_t)b * Cin * N_;
  v8f acc = {};
  if (ALIGNED) {
    const float* ar = xb + row0 + l16;
    const float* bc = xb + col0 + l16;
    int offA0 = (hi ? 2 : 0) * N_, offA1 = (hi ? 3 : 1) * N_;
    int offB0 = (hi ? 1 : 0) * N_, offB1 = (hi ? 3 : 2) * N_;
    for (int cb = 0; cb < Cin; cb += 4) {
      size_t base = (size_t)cb * N_;
      v2f a, bo;
      a.x  = ar[base + offA0];
      a.y  = ar[base + offA1];
      bo.x = bc[base + offB0];
      bo.y = bc[base + offB1];
      acc = wmma_f32_4(a, bo, acc);
    }
  } else {
    int Cpad = (Cin + 3) & ~3;
    for (int cb = 0; cb < Cpad; cb += 4) {
      int ax = cb + (hi ? 2 : 0), ay = cb + (hi ? 3 : 1);
      int bx = cb + (hi ? 1 : 0), by = cb + (hi ? 3 : 2);
      v2f a, bo;
      a.x  = (ax < Cin) ? xb[(size_t)ax * N_ + row0 + l16] : 0.f;
      a.y  = (ay < Cin) ? xb[(size_t)ay * N_ + row0 + l16] : 0.f;
      bo.x = (bx < Cin) ? xb[(size_t)bx * N_ + col0 + l16] : 0.f;
      bo.y = (by < Cin) ? xb[(size_t)by * N_ + col0 + l16] : 0.f;
      acc = wmma_f32_4(a, bo, acc);
    }
  }
  float sqc = sq[(size_t)b * N_ + col0 + l16];
#pragma unroll
  for (int r = 0; r < 8; ++r) {
    int row = row0 + r + (hi ? 8 : 0);
    sdist[(size_t)row * N_ + col0 + l16] = sqc - 2.f * acc[r];
  }
}

// ---------------------------------------------------------------------------
// Top-KNN smallest per row. One wave32 per row: per-lane sorted 20-list over
// 64 strided elements, then 20-round cross-lane merge (shfl_xor min + ballot).
// ---------------------------------------------------------------------------
__global__ void topk_kernel(const float* __restrict__ sdist, int* __restrict__ knn, int b) {
  int row  = blockIdx.x;
  int lane = threadIdx.x;
  float vals[KNN]; int ids[KNN];
#pragma unroll
  for (int t = 0; t < KNN; ++t) { vals[t] = INF_; ids[t] = 0; }
  const float* rp = sdist + (size_t)row * N_;
  for (int j = 0; j < N_ / 32; ++j) {
    int m = j * 32 + lane;
    float v = rp[m];
    if (v < vals[KNN - 1]) {
      vals[KNN - 1] = v; ids[KNN - 1] = m;
#pragma unroll
      for (int t = KNN - 1; t > 0; --t) {
        if (vals[t] < vals[t - 1]) {
          float tv = vals[t]; vals[t] = vals[t - 1]; vals[t - 1] = tv;
          int   ti = ids[t];  ids[t]  = ids[t - 1];  ids[t - 1]  = ti;
        }
      }
    }
  }
  float head = vals[0]; int headi = ids[0];
  int* out = knn + ((size_t)b * N_ + row) * KNN;
  for (int r = 0; r < KNN; ++r) {
    float mn = head;
#pragma unroll
    for (int off = 16; off >= 1; off >>= 1) mn = fminf(mn, __shfl_xor(mn, off, 32));
    unsigned long long ball = __ballot(head == mn);
    int owner = __ffsll(ball) - 1;
    int widx  = __shfl(headi, owner, 32);
    if (lane == 0) out[r] = widx;
    if (lane == owner) {
#pragma unroll
      for (int t = 0; t < KNN - 1; ++t) { vals[t] = vals[t + 1]; ids[t] = ids[t + 1]; }
      vals[KNN - 1] = INF_; ids[KNN - 1] = 0;
      head = vals[0]; headi = ids[0];
    }
  }
}

// ---------------------------------------------------------------------------
// Edge conv: h[b,co,n,kk] = sum_ci W[co,ci] * e[ci,n,kk]
//   e = [central(x[:,n]) ; x[:,knn]-x[:,n]]
// kk padded 20->32 so each 16-col WMMA tile maps to exactly one point n.
// Outputs: sel[b,co,n] = max_k h (if gamma>=0) else min_k h  (BN/ReLU monotone)
//          per-channel sum/sumsq of h over (b,n,k) via atomics.
// ALIGNED: Cin % 4 == 0 -> branch-free split central/diff K loops, v2f W loads.
// ---------------------------------------------------------------------------
__device__ inline float wval(const float* __restrict__ W, int co, int cip, int Cin, int Cpad) {
  if (cip < Cpad) return (cip < Cin) ? W[(size_t)co * 2 * Cin + cip] : 0.f;
  int c = cip - Cpad;
  return (c < Cin) ? W[(size_t)co * 2 * Cin + Cin + c] : 0.f;
}
__device__ inline float eval_(const float* __restrict__ xb, int cip, int Cin, int Cpad,
                              int n, int m) {
  if (cip < Cpad) return (cip < Cin) ? xb[(size_t)cip * N_ + n] : 0.f;
  int c = cip - Cpad;
  if (c >= Cin) return 0.f;
  return xb[(size_t)c * N_ + m] - xb[(size_t)c * N_ + n];
}

template <bool ALIGNED>
__global__ void edgeconv_kernel(const float* __restrict__ x, const int* __restrict__ knn,
                                const float* __restrict__ W, const float* __restrict__ gamma,
                                float* __restrict__ sel, float* __restrict__ ssum,
                                float* __restrict__ ssq, int Cin, int Cout) {
  int lane = threadIdx.x, l16 = lane & 15;
  bool hi = lane >= 16;
  int co0 = blockIdx.y * 16;
  int b   = blockIdx.z;
  int Cpad = (Cin + 3) & ~3;
  const float* xb = x + (size_t)b * Cin * N_;
  const float* Wrow = W + (size_t)(co0 + l16) * 2 * Cin;   // A-operand row for this lane

  float accS[8], accQ[8], mx[8];
#pragma unroll
  for (int r = 0; r < 8; ++r) {
    accS[r] = 0.f; accQ[r] = 0.f;
    mx[r] = (gamma[co0 + r + (hi ? 8 : 0)] >= 0.f) ? 1.f : -1.f;
  }

  int offB0 = (hi ? 1 : 0), offB1 = (hi ? 3 : 2);
  int offA  = (hi ? 2 : 0);

  for (int t = 0; t < NPW; ++t) {
    int n = blockIdx.x * NPW + t;
    const int* kn = knn + ((size_t)b * N_ + n) * KNN;
    int kk1 = 16 + l16;
    int m0 = kn[l16];                       // half 0: kk = l16 (all valid)
    int m1 = (kk1 < KNN) ? kn[kk1] : n;     // half 1: kk = 16+l16 (valid if < 20)
    bool valid1 = (l16 < KNN - 16);
    v8f d0 = {}, d1 = {};

    if (ALIGNED) {
      // --- central half: B operand independent of kk -> shared by d0/d1 ---
      for (int cb = 0; cb < Cin; cb += 4) {
        v2f a = *(const v2f*)&Wrow[cb + offA];
        v2f bc;
        bc.x = xb[(size_t)(cb + offB0) * N_ + n];
        bc.y = xb[(size_t)(cb + offB1) * N_ + n];
        d0 = wmma_f32_4(a, bc, d0);
        d1 = wmma_f32_4(a, bc, d1);
      }
      // --- difference half ---
      for (int cb = 0; cb < Cin; cb += 4) {
        v2f a = *(const v2f*)&Wrow[Cin + cb + offA];
        size_t r0 = (size_t)(cb + offB0) * N_;
        size_t r1 = (size_t)(cb + offB1) * N_;
        float xn0 = xb[r0 + n], xn1 = xb[r1 + n];
        v2f b0, b1;
        b0.x = xb[r0 + m0] - xn0;  b0.y = xb[r1 + m0] - xn1;
        b1.x = xb[r0 + m1] - xn0;  b1.y = xb[r1 + m1] - xn1;
        d0 = wmma_f32_4(a, b0, d0);
        d1 = wmma_f32_4(a, b1, d1);
      }
    } else {
      int Kdim = 2 * Cpad;
      for (int cb = 0; cb < Kdim; cb += 4) {
        v2f a, b0, b1;
        a.x = wval(W, co0 + l16, cb + (hi ? 2 : 0), Cin, Cpad);
        a.y = wval(W, co0 + l16, cb + (hi ? 3 : 1), Cin, Cpad);
        int ci0 = cb + (hi ? 1 : 0), ci2 = cb + (hi ? 3 : 2);
        b0.x = eval_(xb, ci0, Cin, Cpad, n, m0);
        b0.y = eval_(xb, ci2, Cin, Cpad, n, m0);
        b1.x = eval_(xb, ci0, Cin, Cpad, n, m1);
        b1.y = eval_(xb, ci2, Cin, Cpad, n, m1);
        d0 = wmma_f32_4(a, b0, d0);
        d1 = wmma_f32_4(a, b1, d1);
      }
    }

#pragma unroll
    for (int r = 0; r < 8; ++r) {
      float h0 = d0[r], h1 = d1[r];
      float e  = mx[r] * h0;
      float e1 = valid1 ? (mx[r] * h1) : -INF_;
      e = fmaxf(e, e1);
      float s = h0 + (valid1 ? h1 : 0.f);
      float q = h0 * h0 + (valid1 ? h1 * h1 : 0.f);
#pragma unroll
      for (int off = 1; off < 16; off <<= 1) {      // reduce within 16-lane half (kk dim)
        e = fmaxf(e, __shfl_xor(e, off, 32));
        s += __shfl_xor(s, off, 32);
        q += __shfl_xor(q, off, 32);
      }
      if (l16 == 0) {
        int co = co0 + r + (hi ? 8 : 0);
        sel[((size_t)b * Cout + co) * N_ + n] = mx[r] * e;   // flip back if min-select
        accS[r] += s; accQ[r] += q;
      }
    }
  }
  if (l16 == 0) {
#pragma unroll
    for (int r = 0; r < 8; ++r) {
      int co = co0 + r + (hi ? 8 : 0);
      atomicAdd(&ssum[co], accS[r]);
      atomicAdd(&ssq[co],  accQ[r]);
    }
  }
}

// ---------------------------------------------------------------------------
// BN finalize + apply
// ---------------------------------------------------------------------------
__global__ void finalize_kernel(const float* __restrict__ ssum, const float* __restrict__ ssq,
                                const float* __restrict__ gamma, const float* __restrict__ beta,
                                float* __restrict__ scale, float* __restrict__ shift,
                                int Cout, float invcnt) {
  int co = blockIdx.x * blockDim.x + threadIdx.x;
  if (co >= Cout) return;
  float mean = ssum[co] * invcnt;
  float var  = ssq[co] * invcnt - mean * mean;
  float sc   = gamma[co] * rsqrtf(var + EPSV);
  scale[co] = sc;
  shift[co] = beta[co] - mean * sc;
}

__global__ void apply_kernel(const float* __restrict__ sel, const float* __restrict__ scale,
                             const float* __restrict__ shift, float* __restrict__ out, int Cout) {
  int i = blockIdx.x * blockDim.x + threadIdx.x;
  int total = B_ * Cout * N_;
  if (i >= total) return;
  int co = (i / N_) % Cout;
  float v = sel[i] * scale[co] + shift[co];
  out[i] = v > 0.f ? v : 0.f;
}

__global__ void zero_kernel(float* __restrict__ p, int n) {
  int i = blockIdx.x * blockDim.x + threadIdx.x;
  if (i < n) p[i] = 0.f;
}

// ---------------------------------------------------------------------------
// Final 1x1 conv: h = Wf(1024x448) @ concat(f1,f2,f3); pre-BN h -> d_out.
// ---------------------------------------------------------------------------
__device__ inline float featval(const float* __restrict__ f1, const float* __restrict__ f2,
                                const float* __restrict__ f3, int b, int ci, int n) {
  if (ci < 64)  return f1[((size_t)b * 64  + ci)        * N_ + n];
  if (ci < 192) return f2[((size_t)b * 128 + (ci - 64)) * N_ + n];
  return               f3[((size_t)b * 256 + (ci - 192))* N_ + n];
}

__global__ void finalgemm_kernel(const float* __restrict__ f1, const float* __restrict__ f2,
                                 const float* __restrict__ f3, const float* __restrict__ Wf,
                                 float* __restrict__ out) {
  int lane = threadIdx.x, l16 = lane & 15;
  bool hi = lane >= 16;
  int col0 = blockIdx.x * 16, co0 = blockIdx.y * 16, b = blockIdx.z;
  int n = col0 + l16;
  const float* Wrow = Wf + (size_t)(co0 + l16) * 448;
  int offA = (hi ? 2 : 0);
  const float* f2b = f2 + (size_t)b * 128 * N_ + n - (size_t)64  * N_;
  const float* f3b = f3 + (size_t)b * 256 * N_ + n - (size_t)192 * N_;
  const float* f1b = f1 + (size_t)b * 64  * N_ + n;
  v8f acc = {};
  // ci in [0,64): f1
  for (int cb = 0; cb < 64; cb += 4) {
    v2f a = *(const v2f*)&Wrow[cb + offA];
    v2f bo;
    bo.x = f1b[(size_t)(cb + (hi ? 1 : 0)) * N_];
    bo.y = f1b[(size_t)(cb + (hi ? 3 : 2)) * N_];
    acc = wmma_f32_4(a, bo, acc);
  }
  // ci in [64,192): f2
  for (int cb = 64; cb < 192; cb += 4) {
    v2f a = *(const v2f*)&Wrow[cb + offA];
    v2f bo;
    bo.x = f2b[(size_t)(cb + (hi ? 1 : 0)) * N_];
    bo.y = f2b[(size_t)(cb + (hi ? 3 : 2)) * N_];
    acc = wmma_f32_4(a, bo, acc);
  }
  // ci in [192,448): f3
  for (int cb = 192; cb < 448; cb += 4) {
    v2f a = *(const v2f*)&Wrow[cb + offA];
    v2f bo;
    bo.x = f3b[(size_t)(cb + (hi ? 1 : 0)) * N_];
    bo.y = f3b[(size_t)(cb + (hi ? 3 : 2)) * N_];
    acc = wmma_f32_4(a, bo, acc);
  }
#pragma unroll
  for (int r = 0; r < 8; ++r) {
    int co = co0 + r + (hi ? 8 : 0);
    out[((size_t)b * 1024 + co) * N_ + n] = acc[r];
  }
}

// Per-channel stats over (B,N) from stored h (deterministic tree reduce).
__global__ void colstats_kernel(const float* __restrict__ h, float* __restrict__ ssum,
                                float* __restrict__ ssq, int Cout) {
  int co = blockIdx.x, tid = threadIdx.x;
  __shared__ float sh_s[256], sh_q[256];
  float s = 0.f, q = 0.f;
  for (int i = tid; i < B_ * N_; i += 256) {
    int b = i / N_, n = i % N_;
    float v = h[((size_t)b * Cout + co) * N_ + n];
    s += v; q += v * v;
  }
  sh_s[tid] = s; sh_q[tid] = q; __syncthreads();
  for (int off = 128; off; off >>= 1) {
    if (tid < off) { sh_s[tid] += sh_s[tid + off]; sh_q[tid] += sh_q[tid + off]; }
    __syncthreads();
  }
  if (tid == 0) { ssum[co] = sh_s[0]; ssq[co] = sh_q[0]; }
}

// ---------------------------------------------------------------------------
extern "C" void kernel_launch(void* const* d_in, const int* in_sizes, int n_in,
                              void* d_out, int out_size, void* d_ws, size_t ws_size,
                              hipStream_t stream) {
  const float* x  = (const float*)d_in[0];
  const float* W1 = (const float*)d_in[1];
  const float* g1 = (const float*)d_in[2];
  const float* b1 = (const float*)d_in[3];
  const float* W2 = (const float*)d_in[4];
  const float* g2 = (const float*)d_in[5];
  const float* b2 = (const float*)d_in[6];
  const float* W3 = (const float*)d_in[7];
  const float* g3 = (const float*)d_in[8];
  const float* b3 = (const float*)d_in[9];
  const float* Wf = (const float*)d_in[10];
  const float* gf = (const float*)d_in[11];
  const float* bf = (const float*)d_in[12];
  (void)in_sizes; (void)n_in; (void)out_size; (void)ws_size;

  float* ws = (float*)d_ws;
  size_t o = 0;
  float* dist  = ws + o; o += (size_t)N_ * N_;            // 16.8 MB, per-batch reuse
  int*   knn   = (int*)(ws + o); o += (size_t)B_ * N_ * KNN;
  float* sq    = ws + o; o += (size_t)B_ * N_;
  float* f1    = ws + o; o += (size_t)B_ * 64  * N_;
  float* f2    = ws + o; o += (size_t)B_ * 128 * N_;
  float* f3    = ws + o; o += (size_t)B_ * 256 * N_;
  float* sel   = ws + o; o += (size_t)B_ * 256 * N_;
  float* ssum  = ws + o; o += 1024;
  float* ssq   = ws + o; o += 1024;
  float* scale = ws + o; o += 1024;
  float* shift = ws + o; o += 1024;

  struct Layer { const float* xin; int Cin; const float* W; const float* g; const float* be;
                 int Cout; float* fout; };
  Layer layers[3] = {
    { x,  3,   W1, g1, b1, 64,  f1 },
    { f1, 64,  W2, g2, b2, 128, f2 },
    { f2, 128, W3, g3, b3, 256, f3 },
  };

  for (int l = 0; l < 3; ++l) {
    Layer& L = layers[l];
    bool aligned = (L.Cin % 4) == 0;
    sq_kernel<<<(B_ * N_ + 255) / 256, 256, 0, stream>>>(L.xin, sq, L.Cin);
    for (int b = 0; b < B_; ++b) {
      if (aligned)
        dist_kernel<true><<<dim3(N_ / 16, N_ / 16), 32, 0, stream>>>(L.xin, sq, dist, b, L.Cin);
      else
        dist_kernel<false><<<dim3(N_ / 16, N_ / 16), 32, 0, stream>>>(L.xin, sq, dist, b, L.Cin);
      topk_kernel<<<N_, 32, 0, stream>>>(dist, knn, b);
    }
    zero_kernel<<<(2048 + 255) / 256, 256, 0, stream>>>(ssum, 2048);  // ssum+ssq contiguous
    if (aligned)
      edgeconv_kernel<true><<<dim3(N_ / NPW, L.Cout / 16, B_), 32, 0, stream>>>(
          L.xin, knn, L.W, L.g, sel, ssum, ssq, L.Cin, L.Cout);
    else
      edgeconv_kernel<false><<<dim3(N_ / NPW, L.Cout / 16, B_), 32, 0, stream>>>(
          L.xin, knn, L.W, L.g, sel, ssum, ssq, L.Cin, L.Cout);
    float invcnt = 1.f / (float)((size_t)B_ * N_ * KNN);
    finalize_kernel<<<(L.Cout + 255) / 256, 256, 0, stream>>>(ssum, ssq, L.g, L.be,
                                                              scale, shift, L.Cout, invcnt);
    int tot = B_ * L.Cout * N_;
    apply_kernel<<<(tot + 255) / 256, 256, 0, stream>>>(sel, scale, shift, L.fout, L.Cout);
  }

  float* out = (float*)d_out;
  finalgemm_kernel<<<dim3(N_ / 16, 1024 / 16, B_), 32, 0, stream>>>(f1, f2, f3, Wf, out);
  colstats_kernel<<<1024, 256, 0, stream>>>(out, ssum, ssq, 1024);
  finalize_kernel<<<4, 256, 0, stream>>>(ssum, ssq, gf, bf, scale, shift, 1024,
                                         1.f / (float)(B_ * N_));
  int tot = B_ * 1024 * N_;
  apply_kernel<<<(tot + 255) / 256, 256, 0, stream>>>(out, scale, shift, out, 1024);
}